// MultiHeadLatentAttentionInference_56014963475109
// MI455X (gfx1250) — compile-verified
//
#include <hip/hip_runtime.h>
#include <hip/hip_bf16.h>
#include <stdint.h>

typedef __attribute__((ext_vector_type(2))) float v2f;
typedef __attribute__((ext_vector_type(4))) float f4;
typedef __attribute__((ext_vector_type(8))) float v8f;

#define B_SZ     8
#define T_Q      4
#define PAST     4092
#define S_TOT    4096
#define D_MODEL  2048
#define N_HEADS  16
#define D_HEAD   128
#define D_ROPE   64
#define NCHUNK   32
#define SCHUNK   (S_TOT / NCHUNK)   // 128 keys per wave -> 8 tiles of 16

// ---------------------------------------------------------------------------
// f32 WMMA GEMM:  C[32, N] = A[32, K] @ W[N, K]^T
// 4 waves per block; wave w covers K-slice [w*K/4, (w+1)*K/4), partials
// reduced through LDS. Trip count is scalar-uniform (niter = K/16) and the
// per-wave slice is folded into pre-loop pointer bases so the inner loop is
// an SALU-counted, pointer-increment loop (keeps loads clustered and WMMAs
// back-to-back). Fragments per CDNA5 ISA 7.12.2 (32-bit 16x4 A):
//   lane = (phase<<4)|r ; A frag = {A[r, k+2p], A[r, k+2p+1]}
//   B frag = {W[n0+r, k+2p], W[n0+r, k+2p+1]}   (B = W^T tile, 4x16)
//   D: lane (p,r), vgpr v  <->  C[v + 8p, n0 + r]
// ---------------------------------------------------------------------------
__global__ void gemm32_wmma_f32(const float* __restrict__ A,
                                const float* __restrict__ W,
                                float* __restrict__ C, int N, int K) {
    const int n0   = blockIdx.x * 16;
    const int lane = threadIdx.x & 31;
    const int wid  = threadIdx.x >> 5;      // 0..3 (wave-uniform)
    const int r    = lane & 15;
    const int ph   = lane >> 4;

    v8f acc0 = {};
    v8f acc1 = {};

    const int koff  = wid * (K >> 2) + 2 * ph;   // start of this wave's slice
    const int niter = K >> 4;                    // uniform across all waves

    const float* pa0 = A + (size_t)(r)      * K + koff;
    const float* pa1 = A + (size_t)(16 + r) * K + koff;
    const float* pw  = W + (size_t)(n0 + r) * K + koff;

#pragma unroll 4
    for (int it = 0; it < niter; ++it) {
        v2f a0 = *(const v2f*)pa0;
        v2f a1 = *(const v2f*)pa1;
        v2f bw = *(const v2f*)pw;
        acc0 = __builtin_amdgcn_wmma_f32_16x16x4_f32(false, a0, false, bw,
                                                     (short)0, acc0, false, false);
        acc1 = __builtin_amdgcn_wmma_f32_16x16x4_f32(false, a1, false, bw,
                                                     (short)0, acc1, false, false);
        pa0 += 4; pa1 += 4; pw += 4;
    }

    __shared__ __align__(16) float red[4][32][16];
#pragma unroll
    for (int v = 0; v < 8; ++v) {
        red[wid][lane][v]     = acc0[v];
        red[wid][lane][8 + v] = acc1[v];
    }
    __syncthreads();

    if (wid == 0) {
#pragma unroll
        for (int v = 0; v < 8; ++v) {
            const float s0 = red[0][lane][v] + red[1][lane][v] +
                             red[2][lane][v] + red[3][lane][v];
            const float s1 = red[0][lane][8 + v] + red[1][lane][8 + v] +
                             red[2][lane][8 + v] + red[3][lane][8 + v];
            C[(size_t)(v + 8 * ph)      * N + n0 + r] = s0;
            C[(size_t)(16 + v + 8 * ph) * N + n0 + r] = s1;
        }
    }
}

// ---------------------------------------------------------------------------
// RoPE (reference quirk: angle uses t = row % T_Q, positions 0..3).
// ---------------------------------------------------------------------------
__global__ void rope_kernel(float* __restrict__ x, int rows, int nvec) {
    const int total = rows * nvec * 32;
    for (int i = blockIdx.x * blockDim.x + threadIdx.x; i < total;
         i += gridDim.x * blockDim.x) {
        const int p   = i & 31;
        const int rv  = i >> 5;
        const int v   = rv % nvec;
        const int row = rv / nvec;
        const int t   = row & (T_Q - 1);
        const float inv = __powf(10000.0f, -(float)(2 * p) / 64.0f);
        float s, c;
        sincosf((float)t * inv, &s, &c);
        float* e = x + ((size_t)row * nvec + v) * 64 + 2 * p;
        const float x0 = e[0], x1 = e[1];
        e[0] = x0 * c - x1 * s;
        e[1] = x0 * s + x1 * c;
    }
}

// ---------------------------------------------------------------------------
// Concat [past ; new]: pure streaming copy -> non-temporal stores (TH=NT)
// so 536 MB of write-once traffic doesn't thrash WGP$/L2.
// ---------------------------------------------------------------------------
__global__ void concat_kernel(const float* __restrict__ past,
                              const float* __restrict__ newv,
                              float* __restrict__ out, int Spast, int T, int W) {
    const int    S      = Spast + T;
    const size_t total4 = (size_t)B_SZ * S * W / 4;
    for (size_t i = blockIdx.x * (size_t)blockDim.x + threadIdx.x; i < total4;
         i += (size_t)gridDim.x * blockDim.x) {
        const size_t e   = i * 4;
        const size_t b   = e / ((size_t)S * W);
        const size_t rem = e - b * (size_t)S * W;
        const size_t s   = rem / W;
        const size_t d   = rem - s * W;
        const f4* src =
            (s < (size_t)Spast)
                ? (const f4*)(past + (b * Spast + s) * W + d)
                : (const f4*)(newv + (b * T + (s - Spast)) * W + d);
        __builtin_nontemporal_store(*src, (f4*)(out + e));
    }
}

// ---------------------------------------------------------------------------
// Async staging of one 16-key tile (KV head slice + rope rows) into LDS via
// the gfx1250 async-to-LDS path (ASYNCcnt). Each B128 op moves a coalesced
// 512B row slice. 24 async ops per tile.
// ---------------------------------------------------------------------------
__device__ __forceinline__ void issue_tile(const float* __restrict__ kvsrc,
                                           const float* __restrict__ krsrc,
                                           uint32_t kv_lds, uint32_t kr_lds,
                                           int lane) {
    for (int i = lane; i < 16 * (D_HEAD / 4); i += 32) {         // 16 instrs
        const float* g = kvsrc + (size_t)(i >> 5) * D_MODEL + ((i & 31) << 2);
        const uint32_t l = kv_lds + (uint32_t)i * 16;
        asm volatile("global_load_async_to_lds_b128 %0, %1, off"
                     :: "v"(l), "v"(g) : "memory");
    }
    for (int i = lane; i < 16 * (D_ROPE / 4); i += 32) {         // 8 instrs
        const float* g = krsrc + (size_t)i * 4;                  // contiguous
        const uint32_t l = kr_lds + (uint32_t)i * 16;
        asm volatile("global_load_async_to_lds_b128 %0, %1, off"
                     :: "v"(l), "v"(g) : "memory");
    }
}

// ---------------------------------------------------------------------------
// Flash-decode MLA attention. One wave per (chunk, head, b) = 4096 waves.
// Double-buffered async LDS staging overlaps the next tile's HBM traffic
// with this tile's 80 WMMAs (48 score + 32 O-update, all f32 16x16x4).
// ---------------------------------------------------------------------------
__global__ void mla_flash_partial(const float* __restrict__ q_c,
                                  const float* __restrict__ q_r,
                                  const float* __restrict__ c_kv,
                                  const float* __restrict__ k_rope,
                                  const float* __restrict__ mask,
                                  float* __restrict__ part) {
    const int c    = blockIdx.x;
    const int h    = blockIdx.y;
    const int b    = blockIdx.z;
    const int lane = threadIdx.x;
    const int r    = lane & 15;
    const int ph   = lane >> 4;

    __shared__ __align__(16) float qs[16][192];
    __shared__ __align__(16) float kvb[2][16][D_HEAD];
    __shared__ __align__(16) float krb[2][16][D_ROPE];
    __shared__ __align__(16) float pl[16][16];

    const uint32_t kvlds[2] = {(uint32_t)(uintptr_t)&kvb[0][0][0],
                               (uint32_t)(uintptr_t)&kvb[1][0][0]};
    const uint32_t krlds[2] = {(uint32_t)(uintptr_t)&krb[0][0][0],
                               (uint32_t)(uintptr_t)&krb[1][0][0]};

    const int s_begin = c * SCHUNK;
    const float* kv_g = c_kv   + ((size_t)b * S_TOT + s_begin) * D_MODEL + h * D_HEAD;
    const float* kr_g = k_rope + ((size_t)b * S_TOT + s_begin) * D_ROPE;

    // Kick off tile 0 while we stage Q.
    issue_tile(kv_g, kr_g, kvlds[0], krlds[0], lane);

    for (int i = lane; i < 16 * 192; i += 32) {
        const int row = i / 192, col = i - row * 192;
        float v = 0.0f;
        if (row < T_Q) {
            v = (col < D_HEAD)
                    ? q_c[(size_t)(b * T_Q + row) * D_MODEL + h * D_HEAD + col]
                    : q_r[(size_t)(b * T_Q + row) * (N_HEADS * D_ROPE) +
                          h * D_ROPE + (col - D_HEAD)];
        }
        qs[row][col] = v;
    }
    __syncthreads();

    v8f zero = {};
    v8f oacc[8];
#pragma unroll
    for (int j = 0; j < 8; ++j) oacc[j] = zero;
    float m_run[8], l_run[8];
#pragma unroll
    for (int v = 0; v < 8; ++v) { m_run[v] = -1e30f; l_run[v] = 0.0f; }

    const float scale = 0.07216878364870323f;  // 1/sqrt(128+64)
    const int   ntile = SCHUNK / 16;

    for (int t = 0; t < ntile; ++t) {
        const int s0  = s_begin + t * 16;
        const int cur = t & 1;

        // Current tile's async ops (and any older ones) must have landed.
        asm volatile("s_wait_asynccnt 0x0" ::: "memory");

        // Prefetch next tile into the other buffer while we compute.
        if (t + 1 < ntile)
            issue_tile(kv_g + (size_t)(t + 1) * 16 * D_MODEL,
                       kr_g + (size_t)(t + 1) * 16 * D_ROPE,
                       kvlds[cur ^ 1], krlds[cur ^ 1], lane);

        const float (*kv)[D_HEAD] = kvb[cur];
        const float (*kr)[D_ROPE] = krb[cur];

        // scores[16q x 16key] = Q(16x192) @ Kmat^T : 48 WMMAs.
        v8f sacc = {};
#pragma unroll 4
        for (int k = 0; k < 192; k += 4) {
            const int kk = k + 2 * ph;
            v2f a = *(const v2f*)&qs[r][kk];
            const float* bp = (kk < 128) ? &kv[r][kk] : &kr[r][kk - 128];
            v2f bb = *(const v2f*)bp;
            sacc = __builtin_amdgcn_wmma_f32_16x16x4_f32(false, a, false, bb,
                                                         (short)0, sacc, false, false);
        }

        // Online softmax per query row (rows q = v + 8*ph; key column = r).
#pragma unroll
        for (int v = 0; v < 8; ++v) {
            const int q = v + 8 * ph;
            float sc = sacc[v] * scale;
            if (q < T_Q) sc += mask[q * S_TOT + s0 + r];
            float mx = sc;
#pragma unroll
            for (int off = 1; off < 16; off <<= 1)
                mx = fmaxf(mx, __shfl_xor(mx, off));
            const float mnew  = fmaxf(m_run[v], mx);
            const float alpha = __expf(m_run[v] - mnew);
            const float p     = __expf(sc - mnew);
            float rs = p;
#pragma unroll
            for (int off = 1; off < 16; off <<= 1) rs += __shfl_xor(rs, off);
            l_run[v] = l_run[v] * alpha + rs;
            m_run[v] = mnew;
#pragma unroll
            for (int j = 0; j < 8; ++j) oacc[j][v] *= alpha;
            pl[q][r] = p;
        }
        __syncthreads();

        // O[16q x 128d] += P(16x16) @ V(16x128) : 32 WMMAs.
#pragma unroll
        for (int j = 0; j < 8; ++j) {
            const int d0 = j * 16;
#pragma unroll
            for (int k4 = 0; k4 < 16; k4 += 4) {
                v2f a = *(const v2f*)&pl[r][k4 + 2 * ph];
                v2f bb;
                bb.x = kv[k4 + 2 * ph][d0 + r];
                bb.y = kv[k4 + 2 * ph + 1][d0 + r];
                oacc[j] = __builtin_amdgcn_wmma_f32_16x16x4_f32(false, a, false, bb,
                                                                (short)0, oacc[j],
                                                                false, false);
            }
        }
        __syncthreads();
    }

    // Partial record: m[4], l[4], O[4][128]  (520 floats).
    float* pp = part + (size_t)((b * N_HEADS + h) * NCHUNK + c) * 520;
    if (ph == 0) {
        if (r == 0) {
#pragma unroll
            for (int v = 0; v < T_Q; ++v) { pp[v] = m_run[v]; pp[4 + v] = l_run[v]; }
        }
#pragma unroll
        for (int v = 0; v < T_Q; ++v)
#pragma unroll
            for (int j = 0; j < 8; ++j)
                pp[8 + v * D_HEAD + j * 16 + r] = oacc[j][v];
    }
}

// ---------------------------------------------------------------------------
// Merge NCHUNK partial softmax states per (b,h) -> attn[32, 2048] (b,t,h,d).
// ---------------------------------------------------------------------------
__global__ void mla_merge(const float* __restrict__ part, float* __restrict__ attn) {
    const int h = blockIdx.x, b = blockIdx.y, d = threadIdx.x;  // d < 128
    const float* base = part + (size_t)(b * N_HEADS + h) * NCHUNK * 520;
    for (int t = 0; t < T_Q; ++t) {
        float M = -1e30f;
        for (int c = 0; c < NCHUNK; ++c) M = fmaxf(M, base[c * 520 + t]);
        float L = 0.0f, O = 0.0f;
        for (int c = 0; c < NCHUNK; ++c) {
            const float w = __expf(base[c * 520 + t] - M);
            L += w * base[c * 520 + 4 + t];
            O += w * base[c * 520 + 8 + t * D_HEAD + d];
        }
        attn[(size_t)(b * T_Q + t) * D_MODEL + h * D_HEAD + d] = O / L;
    }
}

// ---------------------------------------------------------------------------
extern "C" void kernel_launch(void* const* d_in, const int* in_sizes, int n_in,
                              void* d_out, int out_size, void* d_ws, size_t ws_size,
                              hipStream_t stream) {
    const float* hs    = (const float*)d_in[0];  // [8,4,2048]
    const float* ckv_p = (const float*)d_in[1];  // [8,4092,2048]
    const float* kr_p  = (const float*)d_in[2];  // [8,4092,64]
    const float* mask  = (const float*)d_in[3];  // [1,1,4,4096]
    const float* W_dq  = (const float*)d_in[4];  // [512,2048]
    const float* W_uq  = (const float*)d_in[5];  // [2048,512]
    const float* W_qr  = (const float*)d_in[6];  // [1024,512]
    const float* W_dkv = (const float*)d_in[7];  // [2048,2048]
    const float* W_kr  = (const float*)d_in[8];  // [64,2048]
    const float* W_o   = (const float*)d_in[9];  // [2048,2048]

    float* out   = (float*)d_out;                              // [32,2048]
    float* ckv_o = out + (size_t)32 * D_MODEL;                 // [8,4096,2048]
    float* kr_o  = ckv_o + (size_t)B_SZ * S_TOT * D_MODEL;     // [8,4096,64]

    float* ws      = (float*)d_ws;
    float* ckv_new = ws;  ws += 32 * D_MODEL;            // [32,2048]
    float* kr_new  = ws;  ws += 32 * D_ROPE;             // [32,64]
    float* c_q     = ws;  ws += 32 * 512;                // [32,512]
    float* q_c     = ws;  ws += 32 * D_MODEL;            // [32,2048]
    float* q_r     = ws;  ws += 32 * N_HEADS * D_ROPE;   // [32,1024]
    float* attn    = ws;  ws += 32 * D_MODEL;            // [32,2048]
    float* part    = ws;                                 // [128, NCHUNK, 520] ~8.5MB

    // Projections (exact f32 via V_WMMA_F32_16X16X4_F32, 4-wave K-split).
    gemm32_wmma_f32<<<D_MODEL / 16, 128, 0, stream>>>(hs,  W_dkv, ckv_new, D_MODEL, D_MODEL);
    gemm32_wmma_f32<<<D_ROPE  / 16, 128, 0, stream>>>(hs,  W_kr,  kr_new,  D_ROPE,  D_MODEL);
    gemm32_wmma_f32<<<512     / 16, 128, 0, stream>>>(hs,  W_dq,  c_q,     512,     D_MODEL);
    gemm32_wmma_f32<<<D_MODEL / 16, 128, 0, stream>>>(c_q, W_uq,  q_c,     D_MODEL, 512);
    gemm32_wmma_f32<<<1024    / 16, 128, 0, stream>>>(c_q, W_qr,  q_r,     1024,    512);

    // RoPE (in place) on k_rope_new and q_r.
    rope_kernel<<<1, 256, 0, stream>>>(kr_new, 32, 1);
    rope_kernel<<<4, 256, 0, stream>>>(q_r,    32, N_HEADS);

    // Concatenated KV-cache outputs (attention reads these; NT stores).
    concat_kernel<<<4096, 256, 0, stream>>>(ckv_p, ckv_new, ckv_o, PAST, T_Q, D_MODEL);
    concat_kernel<<<512,  256, 0, stream>>>(kr_p,  kr_new,  kr_o,  PAST, T_Q, D_ROPE);

    // Flash-decode attention: single pass over the 268 MB KV stream,
    // 4096 waves, async double-buffered LDS staging.
    dim3 fgrid(NCHUNK, N_HEADS, B_SZ);
    mla_flash_partial<<<fgrid, 32, 0, stream>>>(q_c, q_r, ckv_o, kr_o, mask, part);
    mla_merge<<<dim3(N_HEADS, B_SZ), D_HEAD, 0, stream>>>(part, attn);

    // Output projection.
    gemm32_wmma_f32<<<D_MODEL / 16, 128, 0, stream>>>(attn, W_o, out, D_MODEL, D_MODEL);
}